// RoI_48610439856225
// MI455X (gfx1250) — compile-verified
//
#include <hip/hip_runtime.h>
#include <stdint.h>

// ROI max pooling (Fast-RCNN style), gfx1250.
// features: (N=8, C=256, H=50, W=50) f32; rois: (R=256, 5) i32
// out: (R, C, 10, 10) f32
//
// Memory-bound gather (no matmul structure -> WMMA not applicable).
// Strategy: one block per (roi, 32-channel chunk). A single Tensor Data
// Mover descriptor DMAs the roi's (w x h x 32ch) strided window from
// global into LDS (tensor_load_to_lds, TENSORcnt), then 256 threads
// compute per-bin maxes from LDS with fully coalesced output writes.

typedef unsigned int v4u __attribute__((ext_vector_type(4)));
typedef int          v8i __attribute__((ext_vector_type(8)));
typedef int          v4i __attribute__((ext_vector_type(4)));

namespace {
constexpr int Cc    = 256;   // channels
constexpr int Hh    = 50;
constexpr int Ww    = 50;
constexpr int OH    = 10;
constexpr int OW    = 10;
constexpr int CHUNK = 32;            // channels per block
constexpr int TILE  = OH * OW;       // 100 outputs per channel
constexpr int NTHREADS = 256;
}

__global__ __launch_bounds__(NTHREADS)
void roipool_tdm_kernel(const float* __restrict__ features,
                        const int* __restrict__ rois,
                        float* __restrict__ out)
{
    __shared__ float smem[CHUNK * TILE];   // 12.8 KB (window is <= 10x10/channel)

    const int r     = blockIdx.x;
    const int cbase = blockIdx.y * CHUNK;
    const int tid   = threadIdx.x;

    // Replicate reference index math exactly (f32 multiply, truncate).
    const int b  = rois[r * 5 + 0];
    const int x1 = (int)((float)rois[r * 5 + 1] * 0.1f);
    const int y1 = (int)((float)rois[r * 5 + 2] * 0.1f);
    int x2 = (int)((float)rois[r * 5 + 3] * 0.1f); if (x2 > Ww) x2 = Ww;
    int y2 = (int)((float)rois[r * 5 + 4] * 0.1f); if (y2 > Hh) y2 = Hh;
    const int w  = x2 - x1;
    const int h  = y2 - y1;
    const int hw = h * w;

    // ---- Stage (w x h x CHUNK) window into LDS with one TDM descriptor ----
    // TDM compacts the tile in LDS: channel stride = h*w floats, row stride = w.
    if (tid < 32 && hw > 0) {            // one wave issues the DMA (EXEC ignored)
        const uint64_t gaddr =
            (uint64_t)(uintptr_t)(features +
                (((size_t)b * Cc + cbase) * Hh + y1) * (size_t)Ww + x1);
        const unsigned lds_base = (unsigned)(uintptr_t)(&smem[0]); // LDS byte offset

        v4u g0;
        g0.x = 1u;                                        // count=1 (user descriptor)
        g0.y = lds_base;                                  // lds_addr
        g0.z = (unsigned)(gaddr & 0xFFFFFFFFu);           // global_addr[31:0]
        g0.w = (unsigned)((gaddr >> 32) & 0x01FFFFFFu)    // global_addr[56:32]
               | (2u << 30);                              // type=2 ("image")

        v8i g1;
        g1[0] = (int)(2u << 16);          // workgroup_mask=0 | data_size=2 (4 bytes)
        g1[1] = (int)((unsigned)w << 16); // atomic_barrier_addr=0 | tensor_dim0[15:0]
        g1[2] = (int)((unsigned)h << 16); // tensor_dim0[31:16]=0 | tensor_dim1[15:0]
        g1[3] = (int)((unsigned)w << 16); // tensor_dim1[31:16]=0 | tile_dim0=w
        g1[4] = (int)((unsigned)h | ((unsigned)CHUNK << 16)); // tile_dim1=h | tile_dim2=32
        g1[5] = Ww;                       // tensor_dim0_stride[31:0] = 50 (elements)
        g1[6] = (int)(((unsigned)(Hh * Ww) & 0xFFFFu) << 16); // strd0[47:32]=0 | strd1[15:0]=2500
        g1[7] = (int)((unsigned)(Hh * Ww) >> 16);             // tensor_dim1_stride[47:16]

        v4i g2;
        g2[0] = CHUNK;                    // tensor_dim2 = 32
        g2[1] = 0;                        // tensor_dim3 (unused, iterate off)
        g2[2] = 0;                        // tensor_dim2_stride[31:0] (unused: tile_dim3=0)
        g2[3] = 0;                        // strd2 hi | tile_dim3=0 -> 3D tile
        v4i g3 = (v4i)0;                  // dims 4+ unused
        v8i g4 = (v8i)0;                  // VADDR4 slot: unused by HW for tensor ops

        __builtin_amdgcn_tensor_load_to_lds(g0, g1, g2, g3, g4, 0);
        __builtin_amdgcn_s_wait_tensorcnt((short)0);
    }
    __syncthreads();

    // ---- Compute per-bin maxes from LDS; coalesced writes ----
    for (int e = tid; e < CHUNK * TILE; e += NTHREADS) {
        int c   = e / TILE;                // divisions by constants only
        int rem = e - c * TILE;
        int j   = rem / OW;                // output row
        int i   = rem - j * OW;            // output col
        int r0 = (j * h) / OH;
        int r1 = ((j + 1) * h + OH - 1) / OH;
        int c0 = (i * w) / OW;
        int c1 = ((i + 1) * w + OW - 1) / OW;
        float m = -__builtin_inff();       // empty bin -> -inf (matches reference)
        const float* tile = &smem[c * hw];
        for (int y = r0; y < r1; ++y) {
            const float* row = tile + y * w;
            for (int x = c0; x < c1; ++x)
                m = fmaxf(m, row[x]);
        }
        out[((size_t)r * Cc + (cbase + c)) * TILE + rem] = m;
    }
}

extern "C" void kernel_launch(void* const* d_in, const int* in_sizes, int n_in,
                              void* d_out, int out_size, void* d_ws, size_t ws_size,
                              hipStream_t stream)
{
    (void)n_in; (void)out_size; (void)d_ws; (void)ws_size;
    const float* features = (const float*)d_in[0];
    const int*   rois     = (const int*)d_in[1];
    float*       out      = (float*)d_out;

    const int R = in_sizes[1] / 5;               // 256 rois
    dim3 grid(R, Cc / CHUNK);                    // (256, 8) blocks
    roipool_tdm_kernel<<<grid, NTHREADS, 0, stream>>>(features, rois, out);
}